// MotionFgnn_1305670058141
// MI455X (gfx1250) — compile-verified
//
#include <hip/hip_runtime.h>
#include <hip/hip_bf16.h>

// ---------------------------------------------------------------------------
// MotionFgnn on gfx1250: f32 WMMA (v_wmma_f32_16x16x4_f32) for all GEMMs.
//   T = 820 rows (40 nodes + 780 factors), K = 39 neighbors, D = 128, E = 16
//   11 MPNN layers, HIDDEN = {128,128,256,256,512,512,256,256,128,128,128}
// ---------------------------------------------------------------------------

#define T_ROWS   820
#define T_PAD    832          // 52 tiles of 16
#define M_TILES  52
#define KNB      39
#define KNB_PAD  48           // 3 tiles of 16
#define NDIM     128
#define EDIM     16
#define NNODES   40
#define NLAYERS  11
#define CMAX     512

typedef float v2f __attribute__((ext_vector_type(2)));
typedef float v8f __attribute__((ext_vector_type(8)));

__device__ __forceinline__ v8f wmma_f32(v2f a, v2f b, v8f c) {
  // D(16x16,f32) = A(16x4,f32) * B(4x16,f32) + C
  return __builtin_amdgcn_wmma_f32_16x16x4_f32(
      false, a, false, b, (short)0, c, false, false);
}

// --------------------------------------------------------------------------
// Stage 0: nff[0:40] = node_feats ; nff[40:820] = 0.5*(nf[u]+nf[v]); pad = 0
// --------------------------------------------------------------------------
__global__ void build_nff_kernel(const float* __restrict__ nf,
                                 const int* __restrict__ graph,
                                 float* __restrict__ nff) {
  const int row = blockIdx.x;      // 0..831
  const int c   = threadIdx.x;     // 0..127
  float v = 0.0f;
  if (row < NNODES) {
    v = nf[row * NDIM + c];
  } else if (row < T_ROWS) {
    const int u = graph[row * KNB + 0];
    const int w = graph[row * KNB + 1];
    v = 0.5f * (nf[u * NDIM + c] + nf[w * NDIM + c]);
  }
  nff[row * NDIM + c] = v;
}

// --------------------------------------------------------------------------
// Stage 1a (WMMA): AB[t][j]    = sum_d nff[t][d] * W_e[j][d]        (j<16)
//                  AB[t][16+j] = sum_d nff[t][d] * W_e[j][128+d]
// [832,128] x [128,32] -> [820,32]
// --------------------------------------------------------------------------
__global__ void edge_proj_kernel(const float* __restrict__ nff,
                                 const float* __restrict__ We,   // [16][256]
                                 float* __restrict__ AB) {
  const int lane = threadIdx.x & 31;
  const int half = lane >> 4, l16 = lane & 15;
  const int m0 = blockIdx.x * 16;
  const int n0 = blockIdx.y * 16;          // grid.y == 2 -> c in [0,32)
  const int c  = n0 + l16;

  const float* arow = nff + (size_t)(m0 + l16) * NDIM + 2 * half;
  const float* brow = We + (size_t)(c & 15) * (2 * NDIM) + (c >> 4) * NDIM + 2 * half;

  v8f acc = {};
  for (int kb = 0; kb < NDIM; kb += 4) {
    v2f a = *(const v2f*)(arow + kb);
    v2f b = *(const v2f*)(brow + kb);
    acc = wmma_f32(a, b, acc);
  }
#pragma unroll
  for (int r = 0; r < 8; ++r) {
    const int mrow = m0 + 8 * half + r;
    if (mrow < T_ROWS) AB[(size_t)mrow * 32 + c] = acc[r];
  }
}

// --------------------------------------------------------------------------
// Stage 1b: e[t][k][j] = relu(AB[t][j] + AB[graph[t][k]][16+j] + b_e[j]);
//           rows k in [39,48) zero-padded.
// --------------------------------------------------------------------------
__global__ void edge_feat_kernel(const float* __restrict__ AB,
                                 const int* __restrict__ graph,
                                 const float* __restrict__ b_e,
                                 float* __restrict__ e) {
  const int t = blockIdx.x;
  for (int idx = threadIdx.x; idx < KNB_PAD * EDIM; idx += blockDim.x) {
    const int k = idx >> 4, j = idx & 15;
    float val = 0.0f;
    if (k < KNB) {
      const int nb = graph[t * KNB + k];
      const float v = AB[(size_t)t * 32 + j] + AB[(size_t)nb * 32 + 16 + j] + b_e[j];
      val = v > 0.0f ? v : 0.0f;
    }
    e[((size_t)t * KNB_PAD + k) * EDIM + j] = val;
  }
}

// --------------------------------------------------------------------------
// Per-layer dense GEMM (WMMA): out[t][c] = sum_d A[t][d]*W[c][d] + bias[c]
// A: [T_PAD][K] row-major (pad rows finite), W: [N][K], store rows < Mstore.
// One wave per 16x16 output tile.
// --------------------------------------------------------------------------
__global__ void gemm_wmma_kernel(const float* __restrict__ A,
                                 const float* __restrict__ W,
                                 const float* __restrict__ bias,
                                 float* __restrict__ out,
                                 int K, int Nsz, int Mstore) {
  const int lane = threadIdx.x & 31;
  const int half = lane >> 4, l16 = lane & 15;
  const int m0 = blockIdx.x * 16;
  const int n0 = blockIdx.y * 16;

  const float* arow = A + (size_t)(m0 + l16) * K + 2 * half;
  const float* brow = W + (size_t)(n0 + l16) * K + 2 * half;

  v8f acc = {};
  for (int kb = 0; kb < K; kb += 4) {
    v2f a = *(const v2f*)(arow + kb);
    v2f b = *(const v2f*)(brow + kb);
    acc = wmma_f32(a, b, acc);
  }
  const float bv = bias[n0 + l16];
#pragma unroll
  for (int r = 0; r < 8; ++r) {
    const int mrow = m0 + 8 * half + r;
    if (mrow < Mstore) out[(size_t)mrow * Nsz + n0 + l16] = acc[r] + bv;
  }
}

// --------------------------------------------------------------------------
// Fused per-layer q-GEMM + gather-multiply + max over neighbors + relu.
//   Q_t = E_t(48x16) @ Wq^T(16xCout) + bq   (WMMA, 3 M-tiles x 4 K-slices)
//   hout[t][c] = relu( max_{k<39} Q_t[k][c] * g[graph[t][k]][c] )
// Block: 128 threads = 4 waves; each wave owns one 16-wide c tile.
// --------------------------------------------------------------------------
__global__ void __launch_bounds__(128)
layer_fused_kernel(const float* __restrict__ e,     // [820][48][16]
                   const float* __restrict__ Wq,    // [Cout][16]
                   const float* __restrict__ bq,    // [Cout]
                   const float* __restrict__ g,     // [820][Cout]
                   const int* __restrict__ graph,   // [820][39]
                   float* __restrict__ hout,        // [820][Cout]
                   int Cout) {
  const int t    = blockIdx.x;
  const int wave = threadIdx.x >> 5;
  const int lane = threadIdx.x & 31;
  const int half = lane >> 4, l16 = lane & 15;
  const int n0 = (blockIdx.y * 4 + wave) * 16;
  const int c  = n0 + l16;

  const float* brow = Wq + (size_t)c * EDIM + 2 * half;
  const v2f b0 = *(const v2f*)(brow + 0);
  const v2f b1 = *(const v2f*)(brow + 4);
  const v2f b2 = *(const v2f*)(brow + 8);
  const v2f b3 = *(const v2f*)(brow + 12);
  const float bqc = bq[c];

  float m = -3.402823466e38f;
#pragma unroll
  for (int mt = 0; mt < 3; ++mt) {
    const float* arow = e + ((size_t)t * KNB_PAD + mt * 16 + l16) * EDIM + 2 * half;
    const v2f a0 = *(const v2f*)(arow + 0);
    const v2f a1 = *(const v2f*)(arow + 4);
    const v2f a2 = *(const v2f*)(arow + 8);
    const v2f a3 = *(const v2f*)(arow + 12);
    v8f acc = {};
    acc = wmma_f32(a0, b0, acc);
    acc = wmma_f32(a1, b1, acc);
    acc = wmma_f32(a2, b2, acc);
    acc = wmma_f32(a3, b3, acc);
#pragma unroll
    for (int r = 0; r < 8; ++r) {
      const int k = mt * 16 + 8 * half + r;
      if (k < KNB) {
        const int nb = graph[t * KNB + k];
        const float q = acc[r] + bqc;
        const float val = q * g[(size_t)nb * Cout + c];
        m = fmaxf(m, val);
      }
    }
  }
  m = fmaxf(m, __shfl_xor(m, 16, 32));
  if (half == 0) hout[(size_t)t * Cout + c] = fmaxf(m, 0.0f);
}

// --------------------------------------------------------------------------
extern "C" void kernel_launch(void* const* d_in, const int* in_sizes, int n_in,
                              void* d_out, int out_size, void* d_ws, size_t ws_size,
                              hipStream_t stream) {
  (void)in_sizes; (void)out_size; (void)ws_size;

  const float* node_feats = (const float*)d_in[0];
  const int*   graph      = (const int*)d_in[1];
  const float* W_e        = (const float*)d_in[2];
  const float* b_e        = (const float*)d_in[3];

  static const int HID[NLAYERS] = {128, 128, 256, 256, 512, 512, 256, 256, 128, 128, 128};
  const float *Wg[NLAYERS], *bg[NLAYERS], *Wq[NLAYERS], *bq[NLAYERS];

  if (n_in >= 4 + 4 * NLAYERS) {
    // Each list entry is its own flat input: Wg[0..10], bg[0..10], Wq[..], bq[..]
    for (int i = 0; i < NLAYERS; ++i) {
      Wg[i] = (const float*)d_in[4 + i];
      bg[i] = (const float*)d_in[4 + NLAYERS + i];
      Wq[i] = (const float*)d_in[4 + 2 * NLAYERS + i];
      bq[i] = (const float*)d_in[4 + 3 * NLAYERS + i];
    }
  } else {
    // Lists flattened as one concatenated input each.
    const float* wgc = (const float*)d_in[4];
    const float* bgc = (const float*)d_in[5];
    const float* wqc = (const float*)d_in[6];
    const float* bqc = (const float*)d_in[7];
    size_t ow = 0, obg = 0, oq = 0, obq = 0;
    int cin = NDIM;
    for (int i = 0; i < NLAYERS; ++i) {
      const int cout = HID[i];
      Wg[i] = wgc + ow;  ow  += (size_t)cout * cin;
      bg[i] = bgc + obg; obg += cout;
      Wq[i] = wqc + oq;  oq  += (size_t)cout * EDIM;
      bq[i] = bqc + obq; obq += cout;
      cin = cout;
    }
  }

  // Workspace layout (floats)
  float* ws  = (float*)d_ws;
  float* nff = ws;                                   // 832*128
  float* AB  = nff + (size_t)T_PAD * NDIM;           // 832*32
  float* eF  = AB + (size_t)T_PAD * 32;              // 820*48*16
  float* gB  = eF + (size_t)T_ROWS * KNB_PAD * EDIM; // 832*512
  float* h0  = gB + (size_t)T_PAD * CMAX;            // 832*512
  float* h1  = h0 + (size_t)T_PAD * CMAX;            // 832*512
  const size_t ws_used =
      ((size_t)T_PAD * NDIM + (size_t)T_PAD * 32 +
       (size_t)T_ROWS * KNB_PAD * EDIM + 3 * (size_t)T_PAD * CMAX) * sizeof(float);

  // Zero workspace so all padded rows stay finite (capture-safe memset node).
  hipMemsetAsync(d_ws, 0, ws_used, stream);

  // Stage 0/1: joint node+factor features and edge features.
  build_nff_kernel<<<dim3(T_PAD), dim3(NDIM), 0, stream>>>(node_feats, graph, nff);
  edge_proj_kernel<<<dim3(M_TILES, 2), dim3(32), 0, stream>>>(nff, W_e, AB);
  edge_feat_kernel<<<dim3(T_ROWS), dim3(256), 0, stream>>>(AB, graph, b_e, eF);

  // MPNN layers.
  const float* hin = nff;
  int cin = NDIM;
  float* bufs[2] = {h0, h1};
  for (int i = 0; i < NLAYERS; ++i) {
    const int cout = HID[i];
    gemm_wmma_kernel<<<dim3(M_TILES, cout / 16), dim3(32), 0, stream>>>(
        hin, Wg[i], bg[i], gB, cin, cout, T_ROWS);
    float* hout = bufs[i & 1];
    layer_fused_kernel<<<dim3(T_ROWS, cout / 64), dim3(128), 0, stream>>>(
        eF, Wq[i], bq[i], gB, graph, hout, cout);
    hin = hout;
    cin = cout;
  }

  // Output: first 40 rows of final h (Cout==128 -> contiguous 40*128 floats).
  hipMemcpyAsync(d_out, hin, (size_t)NNODES * NDIM * sizeof(float),
                 hipMemcpyDeviceToDevice, stream);
}